// ContrastiveLoss_19232863551511
// MI455X (gfx1250) — compile-verified
//
#include <hip/hip_runtime.h>
#include <hip/hip_bf16.h>

// Problem constants (match reference setup_inputs)
#define B_  32
#define C_  256
#define N_  64
#define S_  64
#define NN_ 4096          // N*N
#define T_INV 10.0f       // 1/T_V = 1/T_Q = 10
#define EPS_ 1e-12f

typedef float v2f __attribute__((ext_vector_type(2)));
typedef float v8f __attribute__((ext_vector_type(8)));

// ---------------- Kernel 0: scatter_idx from num_targets ----------------
__global__ void k_scatter(const int* __restrict__ num_targets, int* __restrict__ scatter) {
  if (threadIdx.x == 0 && blockIdx.x == 0) {
    int idx = 0;
    for (int b = 0; b < B_; ++b) {
      int n = num_targets[b];
      for (int j = 0; j < n && idx < S_; ++j) scatter[idx++] = b;
    }
    for (; idx < S_; ++idx) scatter[idx] = B_ - 1;
  }
}

// ---------------- Kernel 1: normalize sentence features ----------------
__global__ void k_norm_sents(const float* __restrict__ sents, float* __restrict__ sfn) {
  int s = blockIdx.x, c = threadIdx.x;
  float x = sents[s * C_ + c];
  float sq = x * x;
  for (int m = 16; m >= 1; m >>= 1) sq += __shfl_xor(sq, m, 32);
  __shared__ float red[8];
  __shared__ float inv;
  int w = c >> 5, l = c & 31;
  if (l == 0) red[w] = sq;
  __syncthreads();
  if (c == 0) {
    float t = 0.f;
    for (int i = 0; i < 8; ++i) t += red[i];
    inv = 1.0f / fmaxf(sqrtf(t), EPS_);
  }
  __syncthreads();
  sfn[s * C_ + c] = x * inv;
}

// ---------------- Kernel 2: per-sentence argmax IoU over valid (triu) p ----------------
__global__ void k_argmax_iou(const float* __restrict__ iou, int* __restrict__ argmaxp) {
  int s = blockIdx.x;
  float best = -1e30f;
  int bidx = 0x3fffffff;
  for (int p = threadIdx.x; p < NN_; p += 256) {
    int n1 = p >> 6, n2 = p & 63;
    if (n2 >= n1) {
      float v = iou[s * NN_ + p];
      if (v > best || (v == best && p < bidx)) { best = v; bidx = p; }
    }
  }
  __shared__ float bv[256];
  __shared__ int   bi[256];
  bv[threadIdx.x] = best; bi[threadIdx.x] = bidx;
  __syncthreads();
  for (int off = 128; off >= 1; off >>= 1) {
    if ((int)threadIdx.x < off) {
      float v2 = bv[threadIdx.x + off]; int i2 = bi[threadIdx.x + off];
      if (v2 > bv[threadIdx.x] || (v2 == bv[threadIdx.x] && i2 < bi[threadIdx.x])) {
        bv[threadIdx.x] = v2; bi[threadIdx.x] = i2;
      }
    }
    __syncthreads();
  }
  if (threadIdx.x == 0) argmaxp[s] = bi[0];
}

// ---------------- Kernel 3: fused WMMA GEMM + column norm + masked exp-sum ----------------
// Grid: (128 col-chunks, 32 videos), block 256 threads (8 waves).
// Each wave computes one 16x16 D tile: D[s_local, p_local] = sum_k sfn[s,k] * video[b,k,p].
// HBM-bound: video (128 MB) is streamed exactly once; norms + masked exp-sum are fused.
__global__ void k_gemm_negq(const float* __restrict__ video,
                            const float* __restrict__ iou,
                            const float* __restrict__ sfn,
                            const int* __restrict__ scatter,
                            float* __restrict__ partials) {
  const int b     = blockIdx.y;
  const int pbase = blockIdx.x * 32;
  const int tid   = threadIdx.x;
  const int w     = tid >> 5;
  const int lane  = tid & 31;
  const int half  = lane >> 4;       // 0: lanes 0-15, 1: lanes 16-31
  const int lh    = lane & 15;
  const int mr    = w & 3;           // row tile (sentences)
  const int nc    = w >> 2;          // col tile within chunk
  const int rowbase = mr * 16;
  const int p     = pbase + nc * 16 + lh;   // proposal column for this lane

  const float* vb   = video + (size_t)b * C_ * NN_;
  const float* arow = sfn + (rowbase + lh) * C_;

  v8f  acc = {};
  float ss = 0.0f;

  #pragma unroll 8
  for (int k = 0; k < C_; k += 4) {
    const int kk = k + 2 * half;
    // Prefetch the k-block 32 channels ahead (gfx1250 global_prefetch_b8 path).
    {
      const int kp = (k + 32 < C_) ? (k + 32) : (C_ - 4);
      __builtin_prefetch(vb + (size_t)(kp + 2 * half) * NN_ + p, 0, 1);
    }
    // A fragment: A[M=lh, K=kk], A[M=lh, K=kk+1]  (ISA 16x4 f32 layout)
    v2f a = *(const v2f*)(arow + kk);
    // B fragment: B[K=kk, N=lh], B[K=kk+1, N=lh]
    float b0 = vb[(size_t)kk * NN_ + p];
    float b1 = vb[(size_t)(kk + 1) * NN_ + p];
    v2f bf = {b0, b1};
    acc = __builtin_amdgcn_wmma_f32_16x16x4_f32(
        /*neg_a=*/false, a, /*neg_b=*/false, bf,
        /*c_mod=*/(short)0, acc, /*reuse_a=*/false, /*reuse_b=*/false);
    ss += b0 * b0 + b1 * b1;   // column squared-sum (this lane covers half the K)
  }
  // combine the two lane-halves covering the same column p
  ss += __shfl_xor(ss, 16, 32);
  const float inv = 1.0f / fmaxf(sqrtf(ss), EPS_);

  // masked exp-sum per row of this tile
  const int  n1    = p >> 6, n2 = p & 63;
  const bool valid = (n2 >= n1);

  __shared__ float part[8][64];
  for (int i = tid; i < 512; i += 256) ((float*)part)[i] = 0.0f;
  __syncthreads();

  float rowsum[8];
  #pragma unroll
  for (int r = 0; r < 8; ++r) {
    const int s = rowbase + half * 8 + r;      // global sentence index
    const float q = acc[r] * inv;              // normalized score
    const bool is_pos = (b == scatter[s]) && (iou[s * NN_ + p] > 0.5f);
    float term = (valid && !is_pos) ? expf(q * T_INV) : 0.0f;
    // reduce across the 16 lanes of this half (same rows, 16 columns)
    term += __shfl_xor(term, 1, 16);
    term += __shfl_xor(term, 2, 16);
    term += __shfl_xor(term, 4, 16);
    term += __shfl_xor(term, 8, 16);
    rowsum[r] = term;
  }
  if (lh == 0) {
    #pragma unroll
    for (int r = 0; r < 8; ++r) part[w][rowbase + half * 8 + r] = rowsum[r];
  }
  __syncthreads();

  if (tid < S_) {
    float tot = 0.0f;
    #pragma unroll
    for (int i = 0; i < 8; ++i) tot += part[i][tid];
    const int chunk = blockIdx.y * gridDim.x + blockIdx.x;   // 0..4095
    partials[(size_t)tid * 4096 + chunk] = tot;
  }
}

// ---------------- Kernel 4: reduce per-sentence partials ----------------
__global__ void k_reduce_negq(const float* __restrict__ partials, float* __restrict__ negq) {
  int s = blockIdx.x;
  float sum = 0.0f;
  for (int c = threadIdx.x; c < 4096; c += 256) sum += partials[(size_t)s * 4096 + c];
  for (int m = 16; m >= 1; m >>= 1) sum += __shfl_xor(sum, m, 32);
  __shared__ float red[8];
  if ((threadIdx.x & 31) == 0) red[threadIdx.x >> 5] = sum;
  __syncthreads();
  if (threadIdx.x == 0) {
    float t = 0.f;
    for (int i = 0; i < 8; ++i) t += red[i];
    negq[s] = t;
  }
}

// ---------------- Kernel 5: topk-video x all-sentences score matrix ----------------
__global__ void k_topk_scores(const float* __restrict__ video,
                              const float* __restrict__ sfn,
                              const int* __restrict__ scatter,
                              const int* __restrict__ argmaxp,
                              float* __restrict__ allsv) {
  int s = blockIdx.x;
  int b = scatter[s];
  int pstar = argmaxp[s];
  __shared__ float col[C_];
  __shared__ float red[8];
  __shared__ float invs;
  int c = threadIdx.x, w = c >> 5, l = c & 31;
  float x = video[((size_t)b * C_ + c) * NN_ + pstar];
  col[c] = x;
  float sq = x * x;
  for (int m = 16; m >= 1; m >>= 1) sq += __shfl_xor(sq, m, 32);
  if (l == 0) red[w] = sq;
  __syncthreads();
  if (c == 0) {
    float t = 0.f;
    for (int i = 0; i < 8; ++i) t += red[i];
    invs = 1.0f / fmaxf(sqrtf(t), EPS_);
  }
  __syncthreads();
  for (int t = w; t < S_; t += 8) {
    float d = 0.0f;
    for (int i = l; i < C_; i += 32) d += col[i] * sfn[t * C_ + i];
    for (int m = 16; m >= 1; m >>= 1) d += __shfl_xor(d, m, 32);
    if (l == 0) allsv[s * S_ + t] = d * invs;
  }
}

// ---------------- Kernel 6: final two losses ----------------
__global__ void k_final(const float* __restrict__ allsv,
                        const float* __restrict__ negq,
                        float* __restrict__ out) {
  int s = threadIdx.x;   // 64 threads = 2 waves
  float pos = allsv[s * S_ + s];
  float nv = 0.0f;
  for (int t = 0; t < S_; ++t)
    if (t != s) nv += expf(allsv[s * S_ + t] * T_INV);
  float pe = expf(pos * T_INV);
  float lv = -(pos * T_INV - logf(pe + nv));
  float lq = -(pos * T_INV - logf(pe + negq[s]));
  for (int m = 16; m >= 1; m >>= 1) {
    lv += __shfl_xor(lv, m, 32);
    lq += __shfl_xor(lq, m, 32);
  }
  __shared__ float rv[2], rq[2];
  int w = s >> 5, l = s & 31;
  if (l == 0) { rv[w] = lv; rq[w] = lq; }
  __syncthreads();
  if (s == 0) {
    out[0] = (rv[0] + rv[1]) / (float)S_;
    out[1] = (rq[0] + rq[1]) / (float)S_;
  }
}

extern "C" void kernel_launch(void* const* d_in, const int* in_sizes, int n_in,
                              void* d_out, int out_size, void* d_ws, size_t ws_size,
                              hipStream_t stream) {
  (void)in_sizes; (void)n_in; (void)out_size; (void)ws_size;
  const float* video = (const float*)d_in[0];   // [B,C,N,N]
  const float* sents = (const float*)d_in[1];   // [S,C]
  const int*   numt  = (const int*)d_in[2];     // [B]
  const float* iou   = (const float*)d_in[3];   // [S,N,N]
  // d_in[4] = mask2d (triu) — structural, recomputed on device
  float* out = (float*)d_out;

  char* wsb = (char*)d_ws;
  float* sfn      = (float*)(wsb + 0);            // 64*256*4   = 65536 B
  int*   scatter  = (int*)  (wsb + 65536);        // 64*4       = 256 B
  int*   argmaxp  = (int*)  (wsb + 65792);        // 64*4       = 256 B
  float* allsv    = (float*)(wsb + 66048);        // 64*64*4    = 16384 B
  float* partials = (float*)(wsb + 82432);        // 64*4096*4  = 1048576 B
  float* negq     = (float*)(wsb + 1131008);      // 64*4       = 256 B

  k_scatter   <<<1, 32, 0, stream>>>(numt, scatter);
  k_norm_sents<<<S_, C_, 0, stream>>>(sents, sfn);
  k_argmax_iou<<<S_, 256, 0, stream>>>(iou, argmaxp);
  k_gemm_negq <<<dim3(128, B_), 256, 0, stream>>>(video, iou, sfn, scatter, partials);
  k_reduce_negq<<<S_, 256, 0, stream>>>(partials, negq);
  k_topk_scores<<<S_, C_, 0, stream>>>(video, sfn, scatter, argmaxp, allsv);
  k_final     <<<1, S_, 0, stream>>>(allsv, negq, out);
}